// RBF_14843406975707
// MI455X (gfx1250) — compile-verified
//
#include <hip/hip_runtime.h>
#include <math.h>

typedef __attribute__((ext_vector_type(2))) float v2f;
typedef __attribute__((ext_vector_type(8))) float v8f;

// ---- workspace layout (in 32-bit words) ----
#define WS_HIST 0        // 2048 u32 histogram bins
#define WS_CTRL 2048     // control words
#define CTRL_PREFIX   0
#define CTRL_PLEN     1
#define CTRL_REM      2
#define CTRL_RES0     3
#define CTRL_RES1     4
#define CTRL_GAMMA    5
#define WS_XSQ  4096     // N floats
#define WS_YSQ  8192     // M floats

__device__ __forceinline__ unsigned f2ord(float x) {
    unsigned u = __float_as_uint(x);
    return (u & 0x80000000u) ? ~u : (u | 0x80000000u);
}
__device__ __forceinline__ float ord2f(unsigned u) {
    u = (u & 0x80000000u) ? (u ^ 0x80000000u) : ~u;
    return __uint_as_float(u);
}

// ---- row sum-of-squares for X (rows 0..N-1) and Y (rows N..N+M-1) ----
__global__ __launch_bounds__(256) void rowsq_kernel(const float* __restrict__ X,
                                                    const float* __restrict__ Y,
                                                    float* __restrict__ ws_f,
                                                    int N, int M) {
    int r = blockIdx.x * 256 + threadIdx.x;
    if (r >= N + M) return;
    const float* src = (r < N) ? (X + (size_t)r * 64) : (Y + (size_t)(r - N) * 64);
    float s = 0.f;
#pragma unroll
    for (int k = 0; k < 64; ++k) s += src[k] * src[k];
    ws_f[(r < N ? WS_XSQ + r : WS_YSQ + (r - N))] = s;
}

// ---- dnorm2 = -2*X@Y^T + xsq + ysq, one wave per 16x16 tile, f32 WMMA K=4 ----
__global__ __launch_bounds__(128) void gemm_dnorm2_kernel(const float* __restrict__ X,
                                                          const float* __restrict__ Y,
                                                          const float* __restrict__ ws_f,
                                                          float* __restrict__ out,
                                                          int tiles, int tcols, int M) {
    const int lane = threadIdx.x;                 // 0..31 (wave32)
    const int tile = blockIdx.x * 4 + threadIdx.y;
    if (tile >= tiles) return;                    // wave-uniform
    const int trow = tile / tcols;
    const int tcol = tile % tcols;
    const int m     = lane & 15;                  // row-in-tile (A) / col-in-tile (B)
    const int khalf = (lane >> 4) * 2;            // K pair selected by upper half-wave

    const float* __restrict__ xrow = X + (size_t)(trow * 16 + m) * 64;
    const float* __restrict__ yrow = Y + (size_t)(tcol * 16 + m) * 64;

    v8f c = {};
#pragma unroll
    for (int s = 0; s < 16; ++s) {                // K = 64 = 16 steps of K=4
        v2f a, b;
        a.x = xrow[s * 4 + khalf];
        a.y = xrow[s * 4 + khalf + 1];
        b.x = yrow[s * 4 + khalf];
        b.y = yrow[s * 4 + khalf + 1];
        // D = A(16x4,f32) * B(4x16,f32) + C  -> v_wmma_f32_16x16x4_f32
        c = __builtin_amdgcn_wmma_f32_16x16x4_f32(false, a, false, b,
                                                  (short)0, c, false, false);
    }

    const int col   = tcol * 16 + m;
    const float ys  = ws_f[WS_YSQ + col];
    const int rbase = trow * 16 + ((lane >> 4) << 3);
#pragma unroll
    for (int v = 0; v < 8; ++v) {
        const int row = rbase + v;                // C/D layout: VGPR v -> M = v (+8 hi half)
        out[(size_t)row * M + col] = -2.0f * c[v] + ws_f[WS_XSQ + row] + ys;
    }
}

// ---- radix-select machinery for the exact median ----
__global__ void init_rank_kernel(unsigned* ws, unsigned rank) {
    if (threadIdx.x == 0 && blockIdx.x == 0) {
        ws[WS_CTRL + CTRL_PREFIX] = 0u;
        ws[WS_CTRL + CTRL_PLEN]   = 0u;
        ws[WS_CTRL + CTRL_REM]    = rank;
    }
}

__global__ __launch_bounds__(256) void zero_hist_kernel(unsigned* ws) {
    int i = blockIdx.x * 256 + threadIdx.x;
    if (i < 2048) ws[WS_HIST + i] = 0u;
}

__global__ __launch_bounds__(256) void hist_kernel(const float* __restrict__ vals,
                                                   unsigned* __restrict__ ws,
                                                   long long n) {
    __shared__ unsigned lh[2048];
    const unsigned prefix = ws[WS_CTRL + CTRL_PREFIX];
    const unsigned plen   = ws[WS_CTRL + CTRL_PLEN];
    const int bits  = (plen == 0u) ? 11 : ((plen == 11u) ? 11 : 10);
    const int shift = 32 - (int)plen - bits;
    const unsigned nb = 1u << bits;

    for (int i = threadIdx.x; i < 2048; i += 256) lh[i] = 0u;
    __syncthreads();

    long long i = (long long)blockIdx.x * 256 + threadIdx.x;
    const long long stride = (long long)gridDim.x * 256;
    for (; i < n; i += stride) {
        unsigned u = f2ord(vals[i]);
        bool ok = (plen == 0u) || ((u >> (32u - plen)) == prefix);
        if (ok) atomicAdd(&lh[(u >> shift) & (nb - 1u)], 1u);
    }
    __syncthreads();
    for (int j = threadIdx.x; j < (int)nb; j += 256) {
        unsigned cnt = lh[j];
        if (cnt) atomicAdd(&ws[WS_HIST + j], cnt);
    }
}

__global__ void select_kernel(unsigned* ws, int slot) {
    if (threadIdx.x != 0 || blockIdx.x != 0) return;
    unsigned prefix = ws[WS_CTRL + CTRL_PREFIX];
    unsigned plen   = ws[WS_CTRL + CTRL_PLEN];
    unsigned rem    = ws[WS_CTRL + CTRL_REM];
    const int bits  = (plen == 0u) ? 11 : ((plen == 11u) ? 11 : 10);
    const unsigned nb = 1u << bits;
    unsigned cum = 0u, b = 0u;
    for (unsigned i = 0; i < nb; ++i) {
        unsigned c = ws[WS_HIST + i];
        if (cum + c > rem) { b = i; break; }
        cum += c;
    }
    prefix = (prefix << bits) | b;
    plen += (unsigned)bits;
    rem -= cum;
    ws[WS_CTRL + CTRL_PREFIX] = prefix;
    ws[WS_CTRL + CTRL_PLEN]   = plen;
    ws[WS_CTRL + CTRL_REM]    = rem;
    if (plen == 32u) ws[WS_CTRL + CTRL_RES0 + slot] = prefix;  // exact bit pattern
}

__global__ void finalize_gamma_kernel(unsigned* ws, float logN) {
    if (threadIdx.x != 0 || blockIdx.x != 0) return;
    float v1 = ord2f(ws[WS_CTRL + CTRL_RES0]);
    float v2 = ord2f(ws[WS_CTRL + CTRL_RES1]);
    float med = 0.5f * (v1 + v2);
    // h = med / (2 ln N); gamma = 1/(2h) = ln(N) / med
    ws[WS_CTRL + CTRL_GAMMA] = __float_as_uint(logN / med);
}

__global__ __launch_bounds__(256) void exp_kernel(float* __restrict__ out,
                                                  const unsigned* __restrict__ ws,
                                                  long long n) {
    const float gamma = __uint_as_float(ws[WS_CTRL + CTRL_GAMMA]);
    long long i = (long long)blockIdx.x * 256 + threadIdx.x;
    const long long stride = (long long)gridDim.x * 256;
    for (; i < n; i += stride) out[i] = expf(-gamma * out[i]);
}

extern "C" void kernel_launch(void* const* d_in, const int* in_sizes, int n_in,
                              void* d_out, int out_size, void* d_ws, size_t ws_size,
                              hipStream_t stream) {
    const float* X = (const float*)d_in[0];
    const float* Y = (const float*)d_in[1];
    float*    out  = (float*)d_out;
    unsigned* ws   = (unsigned*)d_ws;
    float*    ws_f = (float*)d_ws;

    const int d = 64;
    const int N = in_sizes[0] / d;   // 4096
    const int M = in_sizes[1] / d;   // 4096
    const long long n = (long long)N * M;

    // 1) row norms
    {
        int total = N + M;
        rowsq_kernel<<<(total + 255) / 256, 256, 0, stream>>>(X, Y, ws_f, N, M);
    }

    // 2) dnorm2 via f32 WMMA, one wave per 16x16 tile, 4 waves per block
    {
        const int tcols = M / 16;
        const int tiles = (N / 16) * tcols;
        dim3 block(32, 4);
        gemm_dnorm2_kernel<<<(tiles + 3) / 4, block, 0, stream>>>(X, Y, ws_f, out,
                                                                  tiles, tcols, M);
    }

    // 3) exact median via 32-bit radix select (11+11+10), for both middle ranks
    const unsigned k_lo = (unsigned)(n / 2 - 1);
    const unsigned k_hi = (unsigned)(n / 2);
    const int histBlocks = 2048;   // grid-stride over 16.7M values (L2-resident)
    for (int slot = 0; slot < 2; ++slot) {
        init_rank_kernel<<<1, 32, 0, stream>>>(ws, slot == 0 ? k_lo : k_hi);
        for (int level = 0; level < 3; ++level) {
            zero_hist_kernel<<<8, 256, 0, stream>>>(ws);
            hist_kernel<<<histBlocks, 256, 0, stream>>>(out, ws, n);
            select_kernel<<<1, 32, 0, stream>>>(ws, slot);
        }
    }
    finalize_gamma_kernel<<<1, 32, 0, stream>>>(ws, logf((float)N));

    // 4) K = exp(-gamma * dnorm2), in place
    exp_kernel<<<8192, 256, 0, stream>>>(out, ws, n);
}